// NDP_24223615549736
// MI455X (gfx1250) — compile-verified
//
#include <hip/hip_runtime.h>
#include <cmath>
#include <stdint.h>

#define BDIM      256
#define ADIM      16
#define NACT      50
#define NSTEPS    98          // (NUM_ACTIONS-1) * N_SUB, N_SUB = 2
#define NEVALS    (NSTEPS * 6)

namespace ndp {
// ---- Tsit5 tableau (double) ----
constexpr double dH  = 0.01;
constexpr double A21 = 0.161;
constexpr double A31 = -0.008480655492356989, A32 = 0.335480655492357;
constexpr double A41 = 2.8971530571054935,  A42 = -6.359448489975075,  A43 = 4.3622954328695815;
constexpr double A51 = 5.325864828439257,   A52 = -11.748883564062828, A53 = 7.4955393428898365, A54 = -0.09249506636175525;
constexpr double A61 = 5.86145544294642,    A62 = -12.92096931784711,  A63 = 8.159367898576159,  A64 = -0.071584973281401, A65 = -0.028269050394068383;
constexpr double B1 = 0.09646076681806523, B2 = 0.01, B3 = 0.4798896504144996,
                 B4 = 1.379008574103742,  B5 = -3.290069515436081, B6 = 2.324710524099774;

// p-stage scale factors: dp/dt = -p is linear, so every RK stage value of p
// is a constant multiple of the incoming p (batch-invariant, p0 = 1).
constexpr double S2 = 1.0 - dH * A21;
constexpr double S3 = 1.0 - dH * (A31 + A32 * S2);
constexpr double S4 = 1.0 - dH * (A41 + A42 * S2 + A43 * S3);
constexpr double S5 = 1.0 - dH * (A51 + A52 * S2 + A53 * S3 + A54 * S4);
constexpr double S6 = 1.0 - dH * (A61 + A62 * S2 + A63 * S3 + A64 * S4 + A65 * S5);
constexpr double RSTEP = 1.0 - dH * (B1 + B2 * S2 + B3 * S3 + B4 * S4 + B5 * S5 + B6 * S6);

// ---- composite coefficients (double) ----
constexpr double H2  = dH * dH;
constexpr double HC2d = dH * A21;
constexpr double HC3d = dH * (A31 + A32);
constexpr double HC4d = dH * (A41 + A42 + A43);
constexpr double HC5d = dH * (A51 + A52 + A53 + A54);
constexpr double HC6d = dH * (A61 + A62 + A63 + A64 + A65);
constexpr double HCBd = dH * (B1 + B2 + B3 + B4 + B5 + B6);
constexpr double UA31d = H2 * (A32 * A21);
constexpr double UA41d = H2 * (A42 * A21 + A43 * A31);
constexpr double UA42d = H2 * (A43 * A32);
constexpr double UA51d = H2 * (A52 * A21 + A53 * A31 + A54 * A41);
constexpr double UA52d = H2 * (A53 * A32 + A54 * A42);
constexpr double UA53d = H2 * (A54 * A43);
constexpr double UA61d = H2 * (A62 * A21 + A63 * A31 + A64 * A41 + A65 * A51);
constexpr double UA62d = H2 * (A63 * A32 + A64 * A42 + A65 * A52);
constexpr double UA63d = H2 * (A64 * A43 + A65 * A53);
constexpr double UA64d = H2 * (A65 * A54);
constexpr double HA21d = dH * A21;
constexpr double HA31d = dH * A31, HA32d = dH * A32;
constexpr double HA41d = dH * A41, HA42d = dH * A42, HA43d = dH * A43;
constexpr double HA51d = dH * A51, HA52d = dH * A52, HA53d = dH * A53, HA54d = dH * A54;
constexpr double HA61d = dH * A61, HA62d = dH * A62, HA63d = dH * A63, HA64d = dH * A64, HA65d = dH * A65;

// ---- fully fused fp32 stage recurrence:
//   a_i = q - CUDi*ud + sum_{l<i} G_il*a_l + dot_i*gm,   q = 25*(g - u)
// (a_i is linear in the stage state, so the stage-state construction folds
//  into constants; ALPHA_U=10, BETA=2.5 -> 25 and 10.)
constexpr float CUD1 = 10.0f;
constexpr float CUD2 = float(25.0 * HC2d + 10.0);
constexpr float CUD3 = float(25.0 * HC3d + 10.0);
constexpr float CUD4 = float(25.0 * HC4d + 10.0);
constexpr float CUD5 = float(25.0 * HC5d + 10.0);
constexpr float CUD6 = float(25.0 * HC6d + 10.0);
constexpr float G21 = float(-(10.0 * HA21d));
constexpr float G31 = float(-(25.0 * UA31d + 10.0 * HA31d));
constexpr float G32 = float(-(10.0 * HA32d));
constexpr float G41 = float(-(25.0 * UA41d + 10.0 * HA41d));
constexpr float G42 = float(-(25.0 * UA42d + 10.0 * HA42d));
constexpr float G43 = float(-(10.0 * HA43d));
constexpr float G51 = float(-(25.0 * UA51d + 10.0 * HA51d));
constexpr float G52 = float(-(25.0 * UA52d + 10.0 * HA52d));
constexpr float G53 = float(-(25.0 * UA53d + 10.0 * HA53d));
constexpr float G54 = float(-(10.0 * HA54d));
constexpr float G61 = float(-(25.0 * UA61d + 10.0 * HA61d));
constexpr float G62 = float(-(25.0 * UA62d + 10.0 * HA62d));
constexpr float G63 = float(-(25.0 * UA63d + 10.0 * HA63d));
constexpr float G64 = float(-(25.0 * UA64d + 10.0 * HA64d));
constexpr float G65 = float(-(10.0 * HA65d));
// final update:  u += HCB*ud + sum UB_l*a_l ;  ud += sum HB_l*a_l
constexpr float HCBf = float(HCBd);
constexpr float UB1f = float(H2 * (B2 * A21 + B3 * A31 + B4 * A41 + B5 * A51 + B6 * A61));
constexpr float UB2f = float(H2 * (B3 * A32 + B4 * A42 + B5 * A52 + B6 * A62));
constexpr float UB3f = float(H2 * (B4 * A43 + B5 * A53 + B6 * A63));
constexpr float UB4f = float(H2 * (B5 * A54 + B6 * A64));
constexpr float UB5f = float(H2 * (B6 * A65));
constexpr float HB1f = float(dH * B1), HB2f = float(dH * B2), HB3f = float(dH * B3),
                HB4f = float(dH * B4), HB5f = float(dH * B5), HB6f = float(dH * B6);
} // namespace ndp

// ---------------------------------------------------------------------------
// Setup kernel: phi(p) = psi(p)/sum(psi) * p for all 98 steps x 6 stages.
// Batch-invariant (p0 == 1 for every element), computed ONCE in double into
// d_ws (588 float4 = 9408 bytes), instead of per-block f64 emulation.
// ---------------------------------------------------------------------------
__global__ void ndp_phi_setup(float4* __restrict__ gtab) {
    using namespace ndp;
    const int e = blockIdx.x * blockDim.x + threadIdx.x;
    if (e >= NEVALS) return;
    const int n = e / 6;            // step index
    const int i = e - n * 6;        // stage index 0..5
    // RBF centers exactly as the reference (fp32 linspace + exp)
    const float c1 = expf(-0.33333334f), c2 = expf(-0.6666667f), c3 = expf(-1.0f);
    const double C0 = 1.0, C1 = (double)c1, C2 = (double)c2, C3 = (double)c3;
    const double H0 = 4.0 / C0, H1 = 4.0 / C1, H2d = 4.0 / C2, H3 = 4.0 / C3;
    const double si = (i == 0) ? 1.0 : (i == 1) ? S2 : (i == 2) ? S3
                     : (i == 3) ? S4 : (i == 4) ? S5 : S6;
    const double p  = si * pow(RSTEP, (double)n);
    const double d0 = p - C0, d1 = p - C1, d2 = p - C2, d3 = p - C3;
    const double p0 = exp(-H0 * d0 * d0);
    const double p1 = exp(-H1 * d1 * d1);
    const double p2 = exp(-H2d * d2 * d2);
    const double p3 = exp(-H3 * d3 * d3);
    const double inv = p / (p0 + p1 + p2 + p3);
    gtab[e] = make_float4((float)(p0 * inv), (float)(p1 * inv),
                          (float)(p2 * inv), (float)(p3 * inv));
}

// ---------------------------------------------------------------------------
// Main kernel: one lane per (batch, action); block = 16 batch elements.
// ---------------------------------------------------------------------------
__global__ __launch_bounds__(BDIM) void ndp_flow_kernel(
    const float*  __restrict__ init,   // (B, 32)
    const float*  __restrict__ rbf,    // (B, 64) -> W[b][a][k]
    const float*  __restrict__ goals,  // (B, 16)
    const float4* __restrict__ gtab,   // (588,) phi table in global
    float*        __restrict__ out)    // (B, 50, 16)
{
    using namespace ndp;
    __shared__ float4 sphi[NEVALS];    // 9408 B

    const int tid = threadIdx.x;
    const int b   = blockIdx.x * (BDIM / ADIM) + (tid >> 4);
    const int a   = tid & (ADIM - 1);

    // Early prefetch of per-lane inputs (gfx1250 global_prefetch_b8).
    __builtin_prefetch(&rbf[(size_t)b * 64 + a * 4], 0, 3);
    __builtin_prefetch(&init[(size_t)b * 32 + a], 0, 3);
    __builtin_prefetch(&goals[(size_t)b * ADIM + a], 0, 3);

    // Stage the phi table into LDS with the CDNA5 async global->LDS path:
    // data flows cache->LDS without touching VGPRs, tracked by ASYNCcnt.
    // (low 32 bits of a generic pointer to LDS are the LDS byte offset,
    //  which is exactly the VDST operand of the async load.)
    for (int e = tid; e < NEVALS; e += BDIM) {
        const uint32_t ldsaddr  = (uint32_t)(uintptr_t)(&sphi[e]);
        const float4*  gsrc     = gtab + e;
        asm volatile("global_load_async_to_lds_b128 %0, %1, off"
                     :: "v"(ldsaddr), "v"(gsrc)
                     : "memory");
    }
#if __has_builtin(__builtin_amdgcn_s_wait_asynccnt)
    __builtin_amdgcn_s_wait_asynccnt(0);
#else
    asm volatile("s_wait_asynccnt 0" ::: "memory");
#endif
    __syncthreads();

    // ---- Per-lane constants ----
    const float u0  = init[(size_t)b * 32 + a];
    const float ud0 = init[(size_t)b * 32 + ADIM + a];
    const float g   = goals[(size_t)b * ADIM + a];
    const float4 w  = *reinterpret_cast<const float4*>(rbf + (size_t)b * 64 + a * 4);
    const float gm  = g - u0;          // gmu0 = goals - u0 (fixed at t=0)
    const float gg  = 25.0f * g;

    float u = u0, ud = ud0;
    float* outp = out + (size_t)b * NACT * ADIM + a;
    outp[0] = u;                       // sample 0 = initial u

#define NDP_DOT(P) (w.x * (P).x + w.y * (P).y + w.z * (P).z + w.w * (P).w)

    auto dostep = [&](int n) {
        const float4* __restrict__ ph = &sphi[n * 6];   // uniform -> ds broadcast
        const float q = fmaf(-25.0f, u, gg);            // 25*(g-u)
        const float4 p1v = ph[0], p2v = ph[1], p3v = ph[2],
                     p4v = ph[3], p5v = ph[4], p6v = ph[5];
        const float a1 = fmaf(NDP_DOT(p1v), gm, fmaf(-CUD1, ud, q));
        const float a2 = fmaf(NDP_DOT(p2v), gm,
                          fmaf(G21, a1, fmaf(-CUD2, ud, q)));
        const float a3 = fmaf(NDP_DOT(p3v), gm,
                          fmaf(G32, a2, fmaf(G31, a1, fmaf(-CUD3, ud, q))));
        const float a4 = fmaf(NDP_DOT(p4v), gm,
                          fmaf(G43, a3, fmaf(G42, a2, fmaf(G41, a1,
                          fmaf(-CUD4, ud, q)))));
        const float a5 = fmaf(NDP_DOT(p5v), gm,
                          fmaf(G54, a4, fmaf(G53, a3, fmaf(G52, a2,
                          fmaf(G51, a1, fmaf(-CUD5, ud, q))))));
        const float a6 = fmaf(NDP_DOT(p6v), gm,
                          fmaf(G65, a5, fmaf(G64, a4, fmaf(G63, a3,
                          fmaf(G62, a2, fmaf(G61, a1, fmaf(-CUD6, ud, q)))))));
        u  = fmaf(UB5f, a5, fmaf(UB4f, a4, fmaf(UB3f, a3, fmaf(UB2f, a2,
             fmaf(UB1f, a1, fmaf(HCBf, ud, u))))));
        ud = fmaf(HB6f, a6, fmaf(HB5f, a5, fmaf(HB4f, a4, fmaf(HB3f, a3,
             fmaf(HB2f, a2, fmaf(HB1f, a1, ud))))));
    };
#undef NDP_DOT

    #pragma unroll 1
    for (int s = 1; s < NACT; ++s) {
        dostep(2 * s - 2);
        dostep(2 * s - 1);
        outp[(size_t)s * ADIM] = u;    // sample every N_SUB=2 steps
    }
}

extern "C" void kernel_launch(void* const* d_in, const int* in_sizes, int n_in,
                              void* d_out, int out_size, void* d_ws, size_t ws_size,
                              hipStream_t stream) {
    (void)n_in; (void)out_size; (void)ws_size;
    const float* init  = (const float*)d_in[0];   // (B, 2*16)
    const float* rbf   = (const float*)d_in[1];   // (B, 16*4)
    const float* goals = (const float*)d_in[2];   // (B, 16)
    float*  out  = (float*)d_out;                 // (B, 50, 16)
    float4* gtab = (float4*)d_ws;                 // 588 float4 = 9408 B scratch

    ndp_phi_setup<<<(NEVALS + BDIM - 1) / BDIM, BDIM, 0, stream>>>(gtab);

    const int B = in_sizes[0] / 32;               // 32768
    const int blocks = B / (BDIM / ADIM);         // 16 batch elems / block
    ndp_flow_kernel<<<blocks, BDIM, 0, stream>>>(init, rbf, goals, gtab, out);
}